// InclusiveLoss_71691594105550
// MI455X (gfx1250) — compile-verified
//
#include <hip/hip_runtime.h>
#include <math.h>
#include <stdint.h>

// ---------------------------------------------------------------------------
// InclusiveLoss for MI455X (gfx1250, wave32).
// Memory-bound (~1 GB HBM traffic @ 23.3 TB/s ~= 42us roofline):
//   pass 1: column-wise (batch-dim) logsumexp partials    (328 MB)
//   pass 2: tiny merge -> L[c]
//   pass 3: per-row fused kernel: row LSE + WMMA dot prod (328 MB x + ~328 MB D)
//   pass 4: deterministic scalar reduction
// V_WMMA_F32_16X16X4_F32 (exact f32) is the reduction engine for
//   sum_c D[t,c] * (x[b,c] - L[c]):
// A-tile holds 64 products, B = all-ones 4x16 -> D[m,n] = sum_k A[m,k];
// all 16 output columns are identical, so total = (sum of C-tile)/16,
// independent of the exact VGPR layout. Memory ops are b64/b128 to maximize
// bytes per wave32 memory instruction.
// ---------------------------------------------------------------------------

typedef __attribute__((ext_vector_type(2))) float v2f;
typedef __attribute__((ext_vector_type(8))) float v8f;

#define TPB 256
#define RCHUNK 32   // row chunks for the column-LSE partial pass

// Pass 1: per-column online logsumexp over a chunk of rows.
// Even-C fast path: 2 columns per thread, b64 loads (256B per wave request).
__global__ void col_lse_partial(const float* __restrict__ x, int B, int C,
                                float* __restrict__ partM,
                                float* __restrict__ partS) {
  const int tid = threadIdx.x;
  int rows_per = (B + (int)gridDim.y - 1) / (int)gridDim.y;
  int r0 = blockIdx.y * rows_per;
  int r1 = r0 + rows_per; if (r1 > B) r1 = B;

  if ((C & 1) == 0) {
    int c0 = (blockIdx.x * TPB + tid) * 2;
    if (c0 >= C) return;
    float m0 = -INFINITY, s0 = 0.f, m1 = -INFINITY, s1 = 0.f;
    for (int r = r0; r < r1; ++r) {
      v2f v = *(const v2f*)(x + (size_t)r * (size_t)C + c0);
      float nm0 = fmaxf(m0, v.x);
      s0 = s0 * expf(m0 - nm0) + expf(v.x - nm0); m0 = nm0;
      float nm1 = fmaxf(m1, v.y);
      s1 = s1 * expf(m1 - nm1) + expf(v.y - nm1); m1 = nm1;
    }
    size_t o = (size_t)blockIdx.y * (size_t)C + c0;
    v2f pm; pm.x = m0; pm.y = m1; *(v2f*)(partM + o) = pm;
    v2f ps; ps.x = s0; ps.y = s1; *(v2f*)(partS + o) = ps;
  } else {
    int c = blockIdx.x * TPB + tid;
    if (c >= C) return;
    float m = -INFINITY, s = 0.f;
    for (int r = r0; r < r1; ++r) {
      float v = x[(size_t)r * (size_t)C + c];
      float nm = fmaxf(m, v);
      s = s * expf(m - nm) + expf(v - nm);
      m = nm;
    }
    partM[(size_t)blockIdx.y * (size_t)C + c] = m;
    partS[(size_t)blockIdx.y * (size_t)C + c] = s;
  }
}

// Pass 2: merge RCHUNK partial (m,s) pairs per column -> L[c].
__global__ void col_lse_merge(const float* __restrict__ partM,
                              const float* __restrict__ partS,
                              int C, int nchunk, float* __restrict__ L) {
  int c = blockIdx.x * blockDim.x + threadIdx.x;
  if (c >= C) return;
  float m = -INFINITY;
  for (int k = 0; k < nchunk; ++k)
    m = fmaxf(m, partM[(size_t)k * (size_t)C + c]);
  float s = 0.f;
  for (int k = 0; k < nchunk; ++k) {
    float pm = partM[(size_t)k * (size_t)C + c];
    float ps = partS[(size_t)k * (size_t)C + c];
    s += (ps > 0.f) ? ps * expf(pm - m) : 0.f;   // guard empty chunks
  }
  L[c] = m + logf(s);
}

// Pass 3: one block (8 waves) per row b.
//  phase A: stream row (b128), compute row LSE, store (x - L) into LDS.
//  phase B: WMMA-accumulated dot product sum_c D[t,c]*xrow[c], b64 loads.
__global__ void row_main(const float* __restrict__ x,
                         const float* __restrict__ dist,
                         const long long* __restrict__ target,
                         const float* __restrict__ L, int B, int C,
                         float* __restrict__ cePart,
                         float* __restrict__ incPart) {
  extern __shared__ float smem[];
  float* xrow = smem;          // C floats: holds x[b,c] - L[c]
  float* red  = smem + C;      // 2*TPB floats scratch
  const int b   = blockIdx.x;
  const int tid = threadIdx.x;
  const float* xr = x + (size_t)b * (size_t)C;
  const long long t = target[b];

  // ---- phase A ----
  float m = -INFINITY, s = 0.f;
  if ((C & 3) == 0) {
    const int n4 = C >> 2;
    const float4* xr4   = (const float4*)xr;
    const float4* L4p   = (const float4*)L;
    float4*       xrow4 = (float4*)xrow;
    for (int i = tid; i < n4; i += TPB) {
      float4 v  = xr4[i];
      float4 lv = L4p[i];
      float mx = fmaxf(fmaxf(v.x, v.y), fmaxf(v.z, v.w));
      float nm = fmaxf(m, mx);
      s = s * expf(m - nm) + expf(v.x - nm) + expf(v.y - nm)
                           + expf(v.z - nm) + expf(v.w - nm);
      m = nm;
      float4 o; o.x = v.x - lv.x; o.y = v.y - lv.y;
      o.z = v.z - lv.z; o.w = v.w - lv.w;
      xrow4[i] = o;
    }
  } else {
    for (int c = tid; c < C; c += TPB) {
      float v = xr[c];
      float nm = fmaxf(m, v);
      s = s * expf(m - nm) + expf(v - nm);
      m = nm;
      xrow[c] = v - L[c];
    }
  }
  red[tid] = m; red[TPB + tid] = s;
  __syncthreads();
  for (int off = TPB / 2; off > 0; off >>= 1) {
    if (tid < off) {
      float m1 = red[tid],       s1 = red[TPB + tid];
      float m2 = red[tid + off], s2 = red[TPB + tid + off];
      float nm = fmaxf(m1, m2);
      float ns = ((s1 > 0.f) ? s1 * expf(m1 - nm) : 0.f) +
                 ((s2 > 0.f) ? s2 * expf(m2 - nm) : 0.f);
      red[tid] = nm; red[TPB + tid] = ns;
    }
    __syncthreads();
  }
  const float M = red[0] + logf(red[TPB]);   // row logsumexp

  const float* drow = dist + (size_t)t * (size_t)C;

  // ---- phase B: WMMA f32 reduction (64 products per wave-instruction) ----
  const int lane = tid & 31;
  const int wave = __builtin_amdgcn_readfirstlane(tid >> 5); // scalar -> tight loop
  v8f acc = {};
  v2f bones; bones.x = 1.0f; bones.y = 1.0f;   // all-ones B matrix
  int base = wave * 64;
  if ((C & 1) == 0) {
    const v2f* d2 = (const v2f*)drow;
    const v2f* x2 = (const v2f*)xrow;
    for (; base + 64 <= C; base += 8 * 64) {   // full chunks: no masks at all
      int i = (base >> 1) + lane;
      v2f d  = d2[i];
      v2f xv = x2[i];
      v2f a; a.x = d.x * xv.x; a.y = d.y * xv.y;
      acc = __builtin_amdgcn_wmma_f32_16x16x4_f32(
          false, a, false, bones, (short)0, acc, false, false);
    }
  }
  for (; base < C; base += 8 * 64) {           // tail (<64 elems per wave)
    int c0 = base + lane * 2, c1 = c0 + 1;
    int cc0 = c0 < C ? c0 : C - 1;             // clamp -> in-range loads
    int cc1 = c1 < C ? c1 : C - 1;
    float p0 = drow[cc0] * xrow[cc0];
    float p1 = drow[cc1] * xrow[cc1];
    p0 = (c0 < C) ? p0 : 0.f;                  // v_cndmask, EXEC stays full
    p1 = (c1 < C) ? p1 : 0.f;
    v2f a; a.x = p0; a.y = p1;
    acc = __builtin_amdgcn_wmma_f32_16x16x4_f32(
        false, a, false, bones, (short)0, acc, false, false);
  }
  // All 16 output columns identical -> grand total = (sum of tile)/16.
  float lanesum = 0.f;
#pragma unroll
  for (int i = 0; i < 8; ++i) lanesum += acc[i];
  for (int off = 16; off > 0; off >>= 1)
    lanesum += __shfl_xor(lanesum, off, 32);
  float wtot = lanesum * (1.0f / 16.0f);

  __syncthreads();              // done with red[] as LSE scratch
  if (lane == 0) red[wave] = wtot;
  __syncthreads();
  if (tid == 0) {
    float tot = 0.f;
    for (int w = 0; w < TPB / 32; ++w) tot += red[w];
    incPart[b] = tot;                       // sum_c D[t,c]*logsm_batch[b,c]
    cePart[b]  = (xrow[t] + L[t]) - M;      // x[b,t] - M  (xrow holds x - L)
  }
}

// Pass 4: deterministic fixed-order reduction of per-row partials -> scalar.
__global__ void finalize(const float* __restrict__ cePart,
                         const float* __restrict__ incPart,
                         int B, int C, float* __restrict__ out) {
  __shared__ float sc[TPB], si[TPB];
  int tid = threadIdx.x;
  float ce = 0.f, inc = 0.f;
  for (int b = tid; b < B; b += TPB) { ce += cePart[b]; inc += incPart[b]; }
  sc[tid] = ce; si[tid] = inc;
  __syncthreads();
  for (int off = TPB / 2; off > 0; off >>= 1) {
    if (tid < off) { sc[tid] += sc[tid + off]; si[tid] += si[tid + off]; }
    __syncthreads();
  }
  if (tid == 0) {
    float cev  = -sc[0] / (float)B;
    float incv = -0.5f * si[0] / ((float)C * (float)B);
    out[0] = cev + incv;
  }
}

extern "C" void kernel_launch(void* const* d_in, const int* in_sizes, int n_in,
                              void* d_out, int out_size, void* d_ws, size_t ws_size,
                              hipStream_t stream) {
  const float*     x      = (const float*)d_in[0];
  const float*     dist   = (const float*)d_in[1];
  const long long* target = (const long long*)d_in[2];
  float*           out    = (float*)d_out;

  const int B = in_sizes[2];                        // 8192
  const int C = (int)((long long)in_sizes[0] / B);  // 10000

  // Workspace (floats): partM[RCHUNK*C] | partS[RCHUNK*C] | L[C]
  //                     | cePart[B] | incPart[B]   (~2.7 MB total)
  float* partM   = (float*)d_ws;
  float* partS   = partM + (size_t)RCHUNK * (size_t)C;
  float* L       = partS + (size_t)RCHUNK * (size_t)C;
  float* cePart  = L + C;
  float* incPart = cePart + B;

  const int cols_pt = ((C & 1) == 0) ? 2 : 1;       // must match kernel's path
  dim3 gA((C + TPB * cols_pt - 1) / (TPB * cols_pt), RCHUNK);
  col_lse_partial<<<gA, TPB, 0, stream>>>(x, B, C, partM, partS);
  col_lse_merge<<<(C + TPB - 1) / TPB, TPB, 0, stream>>>(partM, partS, C, RCHUNK, L);

  size_t shmem = (size_t)C * sizeof(float) + 2 * TPB * sizeof(float); // ~42 KB
  row_main<<<B, TPB, shmem, stream>>>(x, dist, target, L, B, C, cePart, incPart);

  finalize<<<1, TPB, 0, stream>>>(cePart, incPart, B, C, out);
}